// GraphClassifier_12000138625366
// MI455X (gfx1250) — compile-verified
//
#include <hip/hip_runtime.h>

// Problem constants from the reference: G=1024 graphs, D=256 features, C=1.
#define GD 1024
#define DD 256

typedef __attribute__((ext_vector_type(2))) float v2f;
typedef __attribute__((ext_vector_type(8))) float v8f;

// ---------------------------------------------------------------------------
// Kernel 1: zero the global per-graph sum accumulator in d_ws (graph-capture
// safe re-initialization every launch).
// ---------------------------------------------------------------------------
__global__ void gc_zero(float* __restrict__ gsums) {
  int i = blockIdx.x * blockDim.x + threadIdx.x;
  if (i < GD) gsums[i] = 0.0f;
}

// ---------------------------------------------------------------------------
// Kernel 2: streaming per-node dot products via V_WMMA_F32_16X16X4_F32,
// segment-summed into LDS, then flushed once per block to global atomics.
//
// Tile = 16 nodes. A-matrix (16x4 f32, ISA layout): lane L (L&15 = row m)
// holds K = {koff, koff+1} where koff = (L>=16) ? 2 : 0.  B-matrix (4x16)
// holds the matching W chunk broadcast to all 16 columns, so after the
// 64-step K loop every column of the 16x16 accumulator equals x_row · W.
// C-layout: lane 0 holds rows 0..7 in c[0..7]; lane 16 holds rows 8..15.
// ---------------------------------------------------------------------------
__global__ __launch_bounds__(256) void gc_pool_dot(
    const float* __restrict__ x, const int* __restrict__ batch,
    const float* __restrict__ W, float* __restrict__ gsums,
    int N, int tilesPerBlock) {
  __shared__ float sW[DD];      // W cached once per block (1 KB)
  __shared__ float sSum[GD];    // per-block segment sums (4 KB)

  const int t = threadIdx.x;
  sW[t] = W[t];                                     // blockDim.x == DD == 256
  for (int g = t; g < GD; g += 256) sSum[g] = 0.0f;
  __syncthreads();

  const int lane = t & 31;
  const int wave = t >> 5;                          // 8 waves per block
  const int numTiles = (N + 15) >> 4;
  const int tile0 = blockIdx.x * tilesPerBlock;
  const int tileEnd = min(tile0 + tilesPerBlock, numTiles);

  const int m    = lane & 15;                       // matrix row owned by lane
  const int koff = (lane >> 4) << 1;                // 0 for lanes 0-15, 2 for 16-31

  for (int tile = tile0 + wave; tile < tileEnd; tile += 8) {
    const int base = tile << 4;
    if (base + 16 <= N) {                           // full tile (wave-uniform)
      const float* xrow = x + (size_t)(base + m) * DD;
      v8f c = {};
      #pragma unroll 4
      for (int kb = 0; kb < DD; kb += 4) {
        v2f a, bm;
        a.x  = xrow[kb + koff];                     // global_load_b64 per lane
        a.y  = xrow[kb + koff + 1];
        bm.x = sW[kb + koff];                       // LDS broadcast of W chunk
        bm.y = sW[kb + koff + 1];
        c = __builtin_amdgcn_wmma_f32_16x16x4_f32(false, a, false, bm,
                                                  (short)0, c, false, false);
      }
      if (m == 0) {                                 // lanes 0 and 16 own results
        const int mbase = base + ((lane >> 4) << 3);
        const int4* bp = (const int4*)(batch + mbase);   // 32B-aligned
        int4 g0 = bp[0], g1 = bp[1];
        atomicAdd(&sSum[g0.x], c[0]);
        atomicAdd(&sSum[g0.y], c[1]);
        atomicAdd(&sSum[g0.z], c[2]);
        atomicAdd(&sSum[g0.w], c[3]);
        atomicAdd(&sSum[g1.x], c[4]);
        atomicAdd(&sSum[g1.y], c[5]);
        atomicAdd(&sSum[g1.z], c[6]);
        atomicAdd(&sSum[g1.w], c[7]);
      }
    } else {                                        // tail tile: scalar path
      const int rem = N - base;
      if (lane < rem) {
        const float* xr = x + (size_t)(base + lane) * DD;
        float d = 0.0f;
        for (int k = 0; k < DD; ++k) d += xr[k] * sW[k];
        atomicAdd(&sSum[batch[base + lane]], d);
      }
    }
  }

  __syncthreads();
  // Flush: sorted batch => only a handful of graphs per block are nonzero.
  for (int g = t; g < GD; g += 256) {
    float v = sSum[g];
    if (v != 0.0f) atomicAdd(&gsums[g], v);
  }
}

// ---------------------------------------------------------------------------
// Kernel 3: counts via binary search on the sorted batch array, then
// mean + bias + sigmoid. 1024 outputs.
// ---------------------------------------------------------------------------
__global__ void gc_finalize(const float* __restrict__ gsums,
                            const int* __restrict__ batch,
                            const float* __restrict__ bias,
                            float* __restrict__ out, int N) {
  int g = blockIdx.x * blockDim.x + threadIdx.x;
  if (g >= GD) return;
  int lo = 0, hi = N;
  while (lo < hi) { int mid = (lo + hi) >> 1; if (batch[mid] <  g) lo = mid + 1; else hi = mid; }
  const int lb = lo;
  hi = N;
  while (lo < hi) { int mid = (lo + hi) >> 1; if (batch[mid] <= g) lo = mid + 1; else hi = mid; }
  const float cnt  = (float)(lo - lb);
  const float mean = gsums[g] / fmaxf(cnt, 1.0f);
  const float z    = mean + bias[0];
  out[g] = 1.0f / (1.0f + expf(-z));
}

// ---------------------------------------------------------------------------
extern "C" void kernel_launch(void* const* d_in, const int* in_sizes, int n_in,
                              void* d_out, int out_size, void* d_ws, size_t ws_size,
                              hipStream_t stream) {
  const float* x     = (const float*)d_in[0];   // [N, 256] f32
  const int*   batch = (const int*)  d_in[1];   // [N] int32 (sorted graph ids)
  const float* W     = (const float*)d_in[2];   // [1, 256] f32
  const float* b     = (const float*)d_in[3];   // [1] f32
  float* out   = (float*)d_out;                 // [1024] f32
  float* gsums = (float*)d_ws;                  // 1024 f32 scratch

  const int N = in_sizes[1];
  const int numTiles = (N + 15) / 16;
  int nBlocks = numTiles < 1024 ? numTiles : 1024;
  if (nBlocks < 1) nBlocks = 1;
  const int tilesPerBlock = (numTiles + nBlocks - 1) / nBlocks;

  gc_zero<<<(GD + 255) / 256, 256, 0, stream>>>(gsums);
  gc_pool_dot<<<nBlocks, 256, 0, stream>>>(x, batch, W, gsums, N, tilesPerBlock);
  gc_finalize<<<(GD + 255) / 256, 256, 0, stream>>>(gsums, batch, b, out, N);
}